// CapsuleNetwork_21715354648938
// MI455X (gfx1250) — compile-verified
//
#include <hip/hip_runtime.h>
#include <hip/hip_bf16.h>

// Problem constants (from reference)
#define BB   128
#define SS   512
#define DD   300
#define CC   300
#define KK   3
#define MM   (BB*SS)           // 65536 rows
#define NPADC 320              // padded N (20 tiles of 16)
#define EPS  1e-8f
#define MASK_NEG_F (-1e9f)

typedef float v2f __attribute__((ext_vector_type(2)));
typedef float v4f __attribute__((ext_vector_type(4)));
typedef float v8f __attribute__((ext_vector_type(8)));

__device__ __forceinline__ float squash_norm_from_sq(float sq) {
    return (sq / (1.f + sq)) * sqrtf(sq / (sq + EPS));
}
__device__ __forceinline__ float squash_fac(float sq) {
    return (sq / (1.f + sq)) * rsqrtf(sq + EPS);
}

// ---------------------------------------------------------------------------
// Kernel 0: repack W_s into zero-padded, K-pair-interleaved layout
//   Wq[kp][c][j] = W_s[2*kp+j][c]  (c < 300, else 0), kp in [0,150), c in [0,320)
// so a wave's B-fragment (K0,K0+1 pair) is one contiguous b64 load.
// ---------------------------------------------------------------------------
__global__ __launch_bounds__(256) void prep_ws_kernel(
    const float* __restrict__ Ws, float* __restrict__ Wq)
{
    int i = blockIdx.x*256 + threadIdx.x;        // over 150*320*2 = 96000
    if (i < (DD/2)*NPADC*2) {
        int j  = i & 1;
        int c  = (i >> 1) % NPADC;
        int kp = (i >> 1) / NPADC;
        Wq[i] = (c < CC) ? Ws[(kp*2 + j)*CC + c] : 0.f;
    }
}

// ---------------------------------------------------------------------------
// Kernel 1: guide capsule squash folded into G[c,k]
// ---------------------------------------------------------------------------
__global__ __launch_bounds__(320) void guide_kernel(
    const float* __restrict__ guide,   // [3,300]
    const float* __restrict__ gw,      // [300,300]
    float* __restrict__ G)             // [300,3]
{
    __shared__ float g[KK*CC];
    __shared__ float fac[KK];
    const int t = threadIdx.x;
    for (int i = t; i < KK*CC; i += 320) g[i] = guide[i];
    __syncthreads();
    if (t == 0) {
        #pragma unroll
        for (int k = 0; k < KK; ++k) {
            float sq = 0.f;
            for (int d = 0; d < CC; ++d) { float v = g[k*CC + d]; sq += v*v; }
            fac[k] = squash_fac(sq);
        }
    }
    __syncthreads();
    if (t < CC) {
        float o0 = 0.f, o1 = 0.f, o2 = 0.f;
        for (int d = 0; d < CC; ++d) {
            float w = gw[t*CC + d];
            o0 += w * g[0*CC + d];
            o1 += w * g[1*CC + d];
            o2 += w * g[2*CC + d];
        }
        G[t*KK + 0] = o0 * fac[0];
        G[t*KK + 1] = o1 * fac[1];
        G[t*KK + 2] = o2 * fac[2];
    }
}

// ---------------------------------------------------------------------------
// Kernel 2: per-batch aspect capsule + t[b,c] = sum_d W_att[c,d]*asp_cap[b,d]
// ---------------------------------------------------------------------------
__global__ __launch_bounds__(320) void aspect_kernel(
    const int*   __restrict__ aspect,
    const float* __restrict__ embed,
    const float* __restrict__ Wa,
    const float* __restrict__ ba,
    const float* __restrict__ Watt,
    float* __restrict__ tvecs)
{
    __shared__ float u[CC];
    __shared__ float zz[CC];
    __shared__ float ac[CC];
    __shared__ float sfac;
    const int b = blockIdx.x, t = threadIdx.x;
    if (t < DD) u[t] = embed[(long)aspect[b]*DD + t];
    __syncthreads();
    if (t < CC) {
        float z = ba[t];
        for (int d = 0; d < DD; ++d) z += u[d] * Wa[d*CC + t];
        zz[t] = z;
    }
    __syncthreads();
    if (t == 0) {
        float sq = 0.f;
        for (int c = 0; c < CC; ++c) sq += zz[c]*zz[c];
        sfac = squash_fac(sq);
    }
    __syncthreads();
    if (t < CC) ac[t] = zz[t] * sfac;
    __syncthreads();
    if (t < CC) {
        float tv = 0.f;
        for (int d = 0; d < CC; ++d) tv += Watt[t*CC + d] * ac[d];
        tvecs[b*CC + t] = tv;
    }
}

// ---------------------------------------------------------------------------
// Kernel 3: main WMMA GEMM  primary = squash((emb*alpha) @ W_s + b_s)
// Block = 32-row M-tile, 5 waves; each wave: 2 row-strips x 4 N-tiles
// (N padded to 320) -> 8 v_wmma_f32_16x16x4_f32 per K-step, branch-free loop.
// A strip staged K-pair-interleaved (v2f) so each a-frag is one ds_load_b64
// landing in an even-aligned VGPR pair (no marshalling moves).
// Fused: squash, primary store, scores = P'.t[b], logits = P'@G.
// ---------------------------------------------------------------------------
#define AP 33   // v2f row-stride per K-pair (2-way-max LDS conflicts)

__global__ __launch_bounds__(160) void primary_gemm_kernel(
    const int*   __restrict__ sentence, // [B*S]
    const float* __restrict__ alpha,    // [B*S]
    const float* __restrict__ embed,    // [V,300]
    const float* __restrict__ Wq,       // [150,320,2] padded/interleaved W_s
    const float* __restrict__ bs,       // [300]
    const float* __restrict__ tvecs,    // [B,300]
    const float* __restrict__ G,        // [300,3]
    float* __restrict__ primary,        // [B*S,300]
    float* __restrict__ scores,         // [B*S]
    float* __restrict__ logits)         // [B*S,3]
{
    __shared__ v2f   Asq[(DD/2)*AP];    // Asq[kp*AP + r] = {A[r][2kp], A[r][2kp+1]}
    __shared__ float P[32*304];         // pre-squash outputs
    __shared__ int   tok[32];
    __shared__ float al[32];
    __shared__ float facs[32];
    __shared__ float s2p[160];
    __shared__ float dsp[160];
    __shared__ float d0p[160];
    __shared__ float d1p[160];
    __shared__ float d2p[160];

    const int tid = threadIdx.x;
    const int m0  = blockIdx.x * 32;
    const int b   = m0 / SS;            // 32 | 512 -> one batch per block

    if (tid < 32) {
        tok[tid] = sentence[m0 + tid];
        al[tid]  = alpha[m0 + tid];
    }
    __syncthreads();

    // Stage alpha-scaled gathered embeddings: b128 global reads (coalesced),
    // K-pair-interleaved b64 LDS writes.
    for (int idx = tid; idx < 32*(DD/4); idx += 160) {
        int r = idx / (DD/4), q4 = idx % (DD/4);
        v4f e = ((const v4f*)(embed + (size_t)tok[r]*DD))[q4];
        float a = al[r];
        v2f p0; p0.x = e.x * a; p0.y = e.y * a;
        v2f p1; p1.x = e.z * a; p1.y = e.w * a;
        Asq[(q4*2 + 0)*AP + r] = p0;
        Asq[(q4*2 + 1)*AP + r] = p1;
    }
    __syncthreads();

    const int lane    = tid & 31;
    const int w       = tid >> 5;          // wave 0..4
    const int rlo     = lane & 15;
    const int khalf   = lane >> 4;         // 0 or 1 (K sub-pair select)
    const int rowbase = khalf * 8;         // C/D rows within strip

    const int col0 = (w*4 + 0)*16 + rlo;
    const int col1 = (w*4 + 1)*16 + rlo;
    const int col2 = (w*4 + 2)*16 + rlo;
    const int col3 = (w*4 + 3)*16 + rlo;

    v8f acc00 = {}, acc01 = {}, acc02 = {}, acc03 = {};
    v8f acc10 = {}, acc11 = {}, acc12 = {}, acc13 = {};

    const v2f* wq = (const v2f*)Wq;

    for (int kk = 0; kk < DD/4; ++kk) {
        const int kp = kk*2 + khalf;       // K-pair index for this half-wave

        const v2f* arow = Asq + (size_t)kp*AP;
        v2f a0 = arow[rlo];                // strip0 rows 0..15
        v2f a1 = arow[16 + rlo];           // strip1 rows 16..31

        const v2f* wrow = wq + (size_t)kp*NPADC;
        v2f b0 = wrow[col0];
        v2f b1 = wrow[col1];
        v2f b2 = wrow[col2];
        v2f b3 = wrow[col3];

        acc00 = __builtin_amdgcn_wmma_f32_16x16x4_f32(false, a0, false, b0, (short)0, acc00, false, false);
        acc01 = __builtin_amdgcn_wmma_f32_16x16x4_f32(false, a0, false, b1, (short)0, acc01, false, false);
        acc02 = __builtin_amdgcn_wmma_f32_16x16x4_f32(false, a0, false, b2, (short)0, acc02, false, false);
        acc03 = __builtin_amdgcn_wmma_f32_16x16x4_f32(false, a0, false, b3, (short)0, acc03, false, false);
        acc10 = __builtin_amdgcn_wmma_f32_16x16x4_f32(false, a1, false, b0, (short)0, acc10, false, false);
        acc11 = __builtin_amdgcn_wmma_f32_16x16x4_f32(false, a1, false, b1, (short)0, acc11, false, false);
        acc12 = __builtin_amdgcn_wmma_f32_16x16x4_f32(false, a1, false, b2, (short)0, acc12, false, false);
        acc13 = __builtin_amdgcn_wmma_f32_16x16x4_f32(false, a1, false, b3, (short)0, acc13, false, false);
    }

    // Spill accumulators (+bias) to LDS (guards only here, outside hot loop).
    {
        v8f* accs0[4] = { &acc00, &acc01, &acc02, &acc03 };
        v8f* accs1[4] = { &acc10, &acc11, &acc12, &acc13 };
        const int cols[4] = { col0, col1, col2, col3 };
        #pragma unroll
        for (int j = 0; j < 4; ++j) {
            const int col = cols[j];
            if (col < CC) {
                const float bias = bs[col];
                #pragma unroll
                for (int i = 0; i < 8; ++i) {
                    P[(rowbase + i)*304 + col]      = (*accs0[j])[i] + bias;
                    P[(16 + rowbase + i)*304 + col] = (*accs1[j])[i] + bias;
                }
            }
        }
    }
    __syncthreads();

    // Row sum-of-squares: 5 deterministic partials per row (32 rows x 5 = 160).
    {
        const int r = tid / 5, q = tid % 5;
        float s2 = 0.f;
        for (int c = q*60; c < q*60 + 60; ++c) { float v = P[r*304 + c]; s2 += v*v; }
        s2p[tid] = s2;
    }
    __syncthreads();
    if (tid < 32) {
        float sq = 0.f;
        for (int q = 0; q < 5; ++q) sq += s2p[tid*5 + q];
        facs[tid] = squash_fac(sq);
    }
    __syncthreads();

    // Squash + stream primary to workspace with b128 stores.
    for (int idx = tid; idx < 32*(DD/4); idx += 160) {
        int r = idx / (DD/4), c4 = idx % (DD/4);
        float f = facs[r];
        v4f v;
        v.x = P[r*304 + c4*4 + 0] * f;
        v.y = P[r*304 + c4*4 + 1] * f;
        v.z = P[r*304 + c4*4 + 2] * f;
        v.w = P[r*304 + c4*4 + 3] * f;
        ((v4f*)(primary + (size_t)(m0 + r)*CC))[c4] = v;
    }

    // Fused row dots: scores (vs t[b]) and 3 guide logits.
    {
        const int r = tid / 5, q = tid % 5;
        const float f = facs[r];
        const float* tv = tvecs + (long)b*CC;
        float ds = 0.f, g0 = 0.f, g1 = 0.f, g2 = 0.f;
        for (int c = q*60; c < q*60 + 60; ++c) {
            float v = P[r*304 + c] * f;
            ds += v * tv[c];
            g0 += v * G[c*KK + 0];
            g1 += v * G[c*KK + 1];
            g2 += v * G[c*KK + 2];
        }
        dsp[tid] = ds; d0p[tid] = g0; d1p[tid] = g1; d2p[tid] = g2;
    }
    __syncthreads();
    if (tid < 32) {
        float ds = 0.f, g0 = 0.f, g1 = 0.f, g2 = 0.f;
        for (int q = 0; q < 5; ++q) {
            ds += dsp[tid*5 + q]; g0 += d0p[tid*5 + q];
            g1 += d1p[tid*5 + q]; g2 += d2p[tid*5 + q];
        }
        scores[m0 + tid] = ds;
        logits[(long)(m0 + tid)*KK + 0] = g0;
        logits[(long)(m0 + tid)*KK + 1] = g1;
        logits[(long)(m0 + tid)*KK + 2] = g2;
    }
}

// ---------------------------------------------------------------------------
// Kernel 4: per-batch masked softmax over S, K-softmax routing, bkc
// contraction, squash-norm output [B,K].
// ---------------------------------------------------------------------------
__global__ __launch_bounds__(512) void routing_kernel(
    const int*   __restrict__ sentence,
    const float* __restrict__ scores,
    const float* __restrict__ logits,
    const float* __restrict__ primary,
    const float* __restrict__ scale_p,
    float* __restrict__ out)
{
    __shared__ float red[512];
    __shared__ float gmS[SS*KK];
    __shared__ float acol[KK*512];

    const int b = blockIdx.x, s = threadIdx.x;
    const long base = (long)b*SS + s;

    float v = (sentence[base] != 0) ? scores[base] : MASK_NEG_F;

    red[s] = v; __syncthreads();
    for (int off = 256; off; off >>= 1) {
        if (s < off) red[s] = fmaxf(red[s], red[s + off]);
        __syncthreads();
    }
    float mx = red[0]; __syncthreads();

    float e = __expf(v - mx);
    red[s] = e; __syncthreads();
    for (int off = 256; off; off >>= 1) {
        if (s < off) red[s] += red[s + off];
        __syncthreads();
    }
    float nw = e / red[0];

    float l0 = logits[base*KK + 0], l1 = logits[base*KK + 1], l2 = logits[base*KK + 2];
    float lm = fmaxf(l0, fmaxf(l1, l2));
    float e0 = __expf(l0 - lm), e1 = __expf(l1 - lm), e2 = __expf(l2 - lm);
    float scl = scale_p[0] * nw / (e0 + e1 + e2);
    gmS[s*KK + 0] = e0 * scl;
    gmS[s*KK + 1] = e1 * scl;
    gmS[s*KK + 2] = e2 * scl;
    __syncthreads();

    float a0 = 0.f, a1 = 0.f, a2 = 0.f;
    if (s < CC) {
        const float* pb = primary + (long)b*SS*CC + s;
        for (int t = 0; t < SS; ++t) {
            float p = pb[(long)t*CC];
            a0 += gmS[t*KK + 0] * p;
            a1 += gmS[t*KK + 1] * p;
            a2 += gmS[t*KK + 2] * p;
        }
    }
    acol[0*512 + s] = (s < CC) ? a0*a0 : 0.f;
    acol[1*512 + s] = (s < CC) ? a1*a1 : 0.f;
    acol[2*512 + s] = (s < CC) ? a2*a2 : 0.f;
    __syncthreads();
    for (int off = 256; off; off >>= 1) {
        if (s < off) {
            acol[0*512 + s] += acol[0*512 + s + off];
            acol[1*512 + s] += acol[1*512 + s + off];
            acol[2*512 + s] += acol[2*512 + s + off];
        }
        __syncthreads();
    }
    if (s < KK) out[b*KK + s] = squash_norm_from_sq(acol[s*512]);
}

// ---------------------------------------------------------------------------
extern "C" void kernel_launch(void* const* d_in, const int* in_sizes, int n_in,
                              void* d_out, int out_size, void* d_ws, size_t ws_size,
                              hipStream_t stream) {
    const int*   sentence = (const int*)  d_in[0];
    const int*   aspect   = (const int*)  d_in[1];
    const float* c_alpha  = (const float*)d_in[2];
    const float* embed    = (const float*)d_in[3];
    const float* W_s      = (const float*)d_in[4];
    const float* b_s      = (const float*)d_in[5];
    const float* W_a      = (const float*)d_in[6];
    const float* b_a      = (const float*)d_in[7];
    const float* W_att    = (const float*)d_in[8];
    const float* guide    = (const float*)d_in[9];
    const float* gweight  = (const float*)d_in[10];
    const float* scale    = (const float*)d_in[11];
    float* out = (float*)d_out;

    float* ws      = (float*)d_ws;
    float* primary = ws;                               // 19,660,800
    float* scores  = primary + (size_t)MM*CC;          // 65,536
    float* logits  = scores  + MM;                     // 196,608
    float* tvecs   = logits  + (size_t)MM*KK;          // 38,400
    float* G       = tvecs   + (size_t)BB*CC;          // 900
    float* Wq      = G       + CC*KK;                  // 96,000 (padded W_s)

    prep_ws_kernel<<<((DD/2)*NPADC*2 + 255)/256, 256, 0, stream>>>(W_s, Wq);
    guide_kernel<<<1, 320, 0, stream>>>(guide, gweight, G);
    aspect_kernel<<<BB, 320, 0, stream>>>(aspect, embed, W_a, b_a, W_att, tvecs);
    primary_gemm_kernel<<<MM/32, 160, 0, stream>>>(sentence, c_alpha, embed, Wq, b_s,
                                                   tvecs, G, primary, scores, logits);
    routing_kernel<<<BB, 512, 0, stream>>>(sentence, scores, logits, primary, scale, out);
}